// AggregateKNN_89352499626123
// MI455X (gfx1250) — compile-verified
//
#include <hip/hip_runtime.h>
#include <hip/hip_bf16.h>

// ---------------------------------------------------------------------------
// AggregateKNN for MI455X (gfx1250, wave32)
//   out[0:128)   = sum over ligand atoms of ligand_atom_feature
//   out[128:384) = mean over ligand atoms of (sum of 16-NN protein features)
// kNN metric per ligand row i: x2_j - 2*y_i.x_j  (monotone in true d2)
// computed with V_WMMA_F32_16X16X4_F32:
//   A(16x4) row = (-2lx, -2ly, -2lz, 1), B(4x16) col = (px, py, pz, |p|^2)
// ---------------------------------------------------------------------------

#define NP 65536
#define NL 4096
#define PF 256
#define LF 128
#define KNN 16

typedef float v2f __attribute__((ext_vector_type(2)));
typedef float v4f __attribute__((ext_vector_type(4)));
typedef float v8f __attribute__((ext_vector_type(8)));

#define TOPK_INSERT(kd, ki, dval, jval)                                      \
  do {                                                                       \
    if ((dval) < kd[15]) {                                                   \
      kd[15] = (dval); ki[15] = (jval);                                      \
      _Pragma("unroll")                                                      \
      for (int _s = 15; _s > 0; --_s) {                                      \
        if (kd[_s] < kd[_s - 1]) {                                           \
          float _td = kd[_s]; kd[_s] = kd[_s - 1]; kd[_s - 1] = _td;         \
          int _ti = ki[_s]; ki[_s] = ki[_s - 1]; ki[_s - 1] = _ti;           \
        }                                                                    \
      }                                                                      \
    }                                                                        \
  } while (0)

// ---- zero the 384-float output ---------------------------------------------
__global__ void zero_out_kernel(float* __restrict__ out) {
  int i = blockIdx.x * blockDim.x + threadIdx.x;
  if (i < LF + PF) out[i] = 0.0f;
}

// ---- ligand context: column sums of [NL, LF] -------------------------------
__global__ __launch_bounds__(LF) void ligand_ctx_kernel(
    const float* __restrict__ lfeat, float* __restrict__ out) {
  const int f = threadIdx.x;
  const int r0 = blockIdx.x * 32;
  float acc = 0.0f;
#pragma unroll 8
  for (int r = 0; r < 32; ++r) {
    acc += lfeat[(size_t)(r0 + r) * LF + f];
  }
  atomicAdd(&out[f], acc);
}

// ---- kNN partial scan (1 wave per block, 32 ligand atoms, protein split) ---
__global__ __launch_bounds__(32) void knn_partial_kernel(
    const float* __restrict__ ppos, const float* __restrict__ lpos,
    float* __restrict__ pdist, int* __restrict__ pidx, int split) {
  const int lane = threadIdx.x;
  const int half = lane >> 4;    // 0: lanes 0-15, 1: lanes 16-31
  const int m = lane & 15;
  const int g = blockIdx.x;      // ligand group of 32 atoms
  const int sp = blockIdx.y;     // protein split index
  const int perSplit = NP / split;
  const int tile0 = (sp * perSplit) >> 4;
  const int ntiles = perSplit >> 4;

  // 32 metric rows; row stride 20 floats = 80B (16B aligned for b128 reads)
  __shared__ __align__(16) float sD[32][20];

  // Build A operands for the two 16-row ligand tiles this wave owns.
  // 16x4 f32 A layout: lanes 0-15 hold (K0,K1), lanes 16-31 hold (K2,K3).
  v2f a0, a1;
  {
    int i0 = g * 32 + m;
    float lx = lpos[3 * i0 + 0], ly = lpos[3 * i0 + 1], lz = lpos[3 * i0 + 2];
    v2f lo0 = {-2.0f * lx, -2.0f * ly};
    v2f hi0 = {-2.0f * lz, 1.0f};
    a0 = half ? hi0 : lo0;
    int i1 = i0 + 16;
    lx = lpos[3 * i1 + 0]; ly = lpos[3 * i1 + 1]; lz = lpos[3 * i1 + 2];
    v2f lo1 = {-2.0f * lx, -2.0f * ly};
    v2f hi1 = {-2.0f * lz, 1.0f};
    a1 = half ? hi1 : lo1;
  }

  float kd[KNN];
  int ki[KNN];
#pragma unroll
  for (int s = 0; s < KNN; ++s) { kd[s] = 3.402823e38f; ki[s] = 0; }

  // Software pipeline: preload tile 0's protein positions.
  float px, py, pz;
  {
    int j0 = (tile0 << 4) + m;
    px = ppos[3 * j0 + 0];
    py = ppos[3 * j0 + 1];
    pz = ppos[3 * j0 + 2];
  }

  for (int t = 0; t < ntiles; ++t) {
    const int tbase = (tile0 + t) << 4;
    const float cx = px, cy = py, cz = pz;

    // Issue next tile's load + prefetch of the tile after, so the global
    // latency hides behind this tile's WMMA + selection work.
    if (t + 1 < ntiles) {
      const int jn = tbase + 16 + m;
      __builtin_prefetch(&ppos[3 * (jn + 16)], 0, 3);
      px = ppos[3 * jn + 0];
      py = ppos[3 * jn + 1];
      pz = ppos[3 * jn + 2];
    }

    const float x2 = cx * cx + cy * cy + cz * cz;
    // B(4x16) f32: lanes 0-15 hold rows (K0,K1) of column n, lanes 16-31 (K2,K3)
    v2f blo = {cx, cy};
    v2f bhi = {cz, x2};
    v2f b = half ? bhi : blo;

    v8f c0 = {};
    v8f c1 = {};
    c0 = __builtin_amdgcn_wmma_f32_16x16x4_f32(false, a0, false, b, (short)0,
                                               c0, false, false);
    c1 = __builtin_amdgcn_wmma_f32_16x16x4_f32(false, a1, false, b, (short)0,
                                               c1, false, false);

    // D layout: VGPR v of lane l holds (row = v + 8*half, col = l&15).
    // Transpose through LDS so each lane owns one ligand row.
#pragma unroll
    for (int v = 0; v < 8; ++v) {
      sD[half * 8 + v][m] = c0[v];
      sD[16 + half * 8 + v][m] = c1[v];
    }
    __syncthreads();

    // Bulk-read this lane's 16-candidate row as 4x b128.
    v4f r0 = *(const v4f*)&sD[lane][0];
    v4f r1 = *(const v4f*)&sD[lane][4];
    v4f r2 = *(const v4f*)&sD[lane][8];
    v4f r3 = *(const v4f*)&sD[lane][12];
    __syncthreads();

#pragma unroll
    for (int c = 0; c < 4; ++c) { TOPK_INSERT(kd, ki, r0[c], tbase + c); }
#pragma unroll
    for (int c = 0; c < 4; ++c) { TOPK_INSERT(kd, ki, r1[c], tbase + 4 + c); }
#pragma unroll
    for (int c = 0; c < 4; ++c) { TOPK_INSERT(kd, ki, r2[c], tbase + 8 + c); }
#pragma unroll
    for (int c = 0; c < 4; ++c) { TOPK_INSERT(kd, ki, r3[c], tbase + 12 + c); }
  }

  const int i = g * 32 + lane;
  const size_t base = ((size_t)i * split + sp) * KNN;
#pragma unroll
  for (int s = 0; s < KNN; ++s) {
    pdist[base + s] = kd[s];
    pidx[base + s] = ki[s];
  }
}

// ---- merge partial top-k lists across splits -------------------------------
__global__ __launch_bounds__(64) void knn_merge_kernel(
    const float* __restrict__ pdist, const int* __restrict__ pidx,
    int* __restrict__ fidx, int split) {
  const int i = blockIdx.x * blockDim.x + threadIdx.x;
  if (i >= NL) return;
  float kd[KNN];
  int ki[KNN];
#pragma unroll
  for (int s = 0; s < KNN; ++s) { kd[s] = 3.402823e38f; ki[s] = 0; }
  const size_t base = (size_t)i * split * KNN;
  const int ncand = split * KNN;
  for (int c = 0; c < ncand; ++c) {
    float d = pdist[base + c];
    int jj = pidx[base + c];
    TOPK_INSERT(kd, ki, d, jj);
  }
#pragma unroll
  for (int s = 0; s < KNN; ++s) fidx[(size_t)i * KNN + s] = ki[s];
}

// ---- gather neighbor features, partial-sum, atomic into protein_ctx --------
__global__ __launch_bounds__(PF) void aggregate_kernel(
    const float* __restrict__ pfeat, const int* __restrict__ fidx,
    float* __restrict__ out) {
  __shared__ int sIdx[256];
  const int f = threadIdx.x;
  sIdx[f] = fidx[(size_t)blockIdx.x * 256 + f];
  __syncthreads();
  float acc = 0.0f;
  for (int r = 0; r < 256; ++r) {
    acc += pfeat[(size_t)sIdx[r] * PF + f];
  }
  atomicAdd(&out[LF + f], acc * (1.0f / (float)NL));
}

// ---------------------------------------------------------------------------
extern "C" void kernel_launch(void* const* d_in, const int* in_sizes, int n_in,
                              void* d_out, int out_size, void* d_ws,
                              size_t ws_size, hipStream_t stream) {
  (void)in_sizes; (void)n_in; (void)out_size;
  const float* ppos = (const float*)d_in[0];   // [NP,3]
  const float* pfeat = (const float*)d_in[1];  // [NP,PF]
  const float* lpos = (const float*)d_in[2];   // [NL,3]
  const float* lfeat = (const float*)d_in[3];  // [NL,LF]
  float* out = (float*)d_out;                  // [LF+PF]

  // Pick the largest protein-axis split whose partial lists fit in d_ws.
  int split = 1;
  for (int s = 32; s >= 1; s >>= 1) {
    size_t need = (size_t)NL * s * KNN * 8 + (size_t)NL * KNN * 4;
    if (need <= ws_size) { split = s; break; }
  }
  float* pdist = (float*)d_ws;
  int* pidx = (int*)(pdist + (size_t)NL * split * KNN);
  int* fidx = (int*)(pidx + (size_t)NL * split * KNN);

  zero_out_kernel<<<(LF + PF + 255) / 256, 256, 0, stream>>>(out);
  ligand_ctx_kernel<<<NL / 32, LF, 0, stream>>>(lfeat, out);
  knn_partial_kernel<<<dim3(NL / 32, split), 32, 0, stream>>>(ppos, lpos,
                                                              pdist, pidx,
                                                              split);
  knn_merge_kernel<<<NL / 64, 64, 0, stream>>>(pdist, pidx, fidx, split);
  aggregate_kernel<<<NL / 16, PF, 0, stream>>>(pfeat, fidx, out);
}